// Integration_4123168604342
// MI455X (gfx1250) — compile-verified
//
#include <hip/hip_runtime.h>
#include <hip/hip_bf16.h>

// x is (T=128, B=4, E=10000, F=16) f32; out is (128, 4, 10000, 15) f32.
#define T_DIM   128
#define B_DIM   4
#define E_DIM   10000
#define F_DIM   16
#define F_OUT   (F_DIM - 1)               // 15
#define BE      (B_DIM * E_DIM)           // 40,000 columns
#define STRIDE_IN  ((size_t)BE * F_DIM)   // input elements per t-step
#define STRIDE_OUT ((size_t)BE * F_OUT)   // output elements per t-step

#define WAVES_PER_BLOCK 8
#define COLS_PER_WAVE   2                 // 2 columns -> 128B contiguous row per t
#define COLS_PER_BLOCK  (WAVES_PER_BLOCK * COLS_PER_WAVE)   // 16
#define NSTAGE  8                         // t-steps buffered per wave (ring)
#define GROUP   4                         // t-steps per async b128 instruction
#define NGROUPS (T_DIM / GROUP)           // 32

typedef __attribute__((address_space(3))) float lds_float_t;

__device__ __forceinline__ float fast_tanh(float v) {
#if __has_builtin(__builtin_amdgcn_tanhf)
    return __builtin_amdgcn_tanhf(v);     // gfx1250 native v_tanh_f32
#else
    return tanhf(v);
#endif
}

#define WAIT_ASYNC(n) asm volatile("s_wait_asynccnt " #n ::: "memory")

__global__ __launch_bounds__(256) void Integration_scan_async_kernel(
    const float* __restrict__ x,
    const int*   __restrict__ dummy_index,
    float*       __restrict__ out)
{
    // 8 waves * 8 slots * (2 cols * 16 ch) floats = 2048 floats = 8KB
    __shared__ __align__(128) float sbuf[WAVES_PER_BLOCK * NSTAGE * COLS_PER_WAVE * F_DIM];

    const int wave = threadIdx.x >> 5;
    const int lane = threadIdx.x & 31;
    const int be0  = (blockIdx.x * WAVES_PER_BLOCK + wave) * COLS_PER_WAVE;

    // ---- async-producer addressing (all 32 lanes participate) ----
    // lane j copies 16B: row t = g*4 + (j>>3), bytes (j&7)*16 within the 128B row
    const int tt   = lane >> 3;           // 0..3  (t within group)
    const int part = lane & 7;            // 0..7  (16B chunk within row)
    const float* gsrc = x + (size_t)be0 * F_DIM + (size_t)tt * STRIDE_IN + part * 4;

    float* wbuf = &sbuf[wave * NSTAGE * COLS_PER_WAVE * F_DIM];   // 256 floats per wave

    // LDS byte offsets (AS3 ptr -> 32-bit LDS address) for even/odd ring halves
    const unsigned lds_lo =
        (unsigned)(size_t)(lds_float_t*)&wbuf[tt * (COLS_PER_WAVE * F_DIM) + part * 4];
    const unsigned lds_hi = lds_lo + GROUP * COLS_PER_WAVE * F_DIM * 4;  // +512B

    auto issue = [&](int g) {
        const float* ga = gsrc + (size_t)g * GROUP * STRIDE_IN;
        const unsigned la = (g & 1) ? lds_hi : lds_lo;
        asm volatile("global_load_async_to_lds_b128 %0, %1, off th:TH_LOAD_NT"
                     :: "v"(la), "v"(ga) : "memory");
    };

    // ---- scan-consumer addressing (lanes 0..29 active) ----
    const bool active = lane < COLS_PER_WAVE * F_OUT;
    const int  cl  = active ? lane : 0;
    const int  col = cl / F_OUT;                 // 0 or 1
    const int  fi  = cl - col * F_OUT;           // 0..14
    const int  d   = dummy_index[0];             // time channel (0 in setup)
    const int  src = fi + (fi >= d ? 1 : 0);     // concat-skip remap

    const float* tcp = wbuf + col * F_DIM + d;   // + slot*32 per t
    const float* fvp = wbuf + col * F_DIM + src;
    float* pout = out + (size_t)(be0 + col) * F_OUT + fi;

    float acc = 0.f, tprev = 0.f;

    auto consume = [&](int g) {
        const float* tg = tcp + ((g & 1) ? GROUP * 32 : 0);
        const float* fg = fvp + ((g & 1) ? GROUP * 32 : 0);
        float* og = pout + (size_t)g * GROUP * STRIDE_OUT;
#pragma unroll
        for (int k = 0; k < GROUP; ++k) {
            const float tc = tg[k * 32];
            const float fv = fg[k * 32];
            acc   = fmaf(fv, tc - tprev, acc);
            tprev = tc;
            if (active)
                __builtin_nontemporal_store(fast_tanh(acc), og + (size_t)k * STRIDE_OUT);
        }
    };

    // ---- pipeline: 2 groups in flight, ring depth 8 t-steps ----
    issue(0);
    issue(1);
    issue(2);
    WAIT_ASYNC(0x2);                     // group 0 resident (completions in order)

    // group 0: peel t=0 (dt0 = t[0] + t[1])
    {
        const float tc0 = tcp[0];
        const float tc1 = tcp[32];
        const float fv0 = fvp[0];
        acc   = fv0 * (tc0 + tc1);
        tprev = tc0;
        if (active) __builtin_nontemporal_store(fast_tanh(acc), pout);
#pragma unroll
        for (int k = 1; k < GROUP; ++k) {
            const float tc = tcp[k * 32];
            const float fv = fvp[k * 32];
            acc   = fmaf(fv, tc - tprev, acc);
            tprev = tc;
            if (active)
                __builtin_nontemporal_store(fast_tanh(acc), pout + (size_t)k * STRIDE_OUT);
        }
    }

    for (int g = 1; g <= NGROUPS - 3; ++g) {     // g = 1..29
        issue(g + 2);
        WAIT_ASYNC(0x2);                          // group g resident
        consume(g);
    }
    WAIT_ASYNC(0x1);                              // group 30 resident
    consume(NGROUPS - 2);
    WAIT_ASYNC(0x0);                              // group 31 resident
    consume(NGROUPS - 1);
}

extern "C" void kernel_launch(void* const* d_in, const int* in_sizes, int n_in,
                              void* d_out, int out_size, void* d_ws, size_t ws_size,
                              hipStream_t stream)
{
    const float* x    = (const float*)d_in[0];
    const int*   didx = (const int*)d_in[1];
    float*       out  = (float*)d_out;

    const int blocks = BE / COLS_PER_BLOCK;       // 2500, exact
    Integration_scan_async_kernel<<<blocks, 256, 0, stream>>>(x, didx, out);
}